// TCLMModel_51324859187424
// MI455X (gfx1250) — compile-verified
//
#include <hip/hip_runtime.h>
#include <hip/hip_bf16.h>
#include <math.h>

// Problem constants (from reference)
#define B_    8
#define E_    2000
#define N_    10000
#define NREL_ 25
#define K_    16
#define T_    3
#define L_    32

typedef __attribute__((ext_vector_type(2))) float v2f;
typedef __attribute__((ext_vector_type(8))) float v8f;

__device__ __forceinline__ float clip01(float x) { return fminf(fmaxf(x, 0.0f), 1.0f); }

// ---------------------------------------------------------------------------
// One-hot index extraction (bandwidth-bound; ~160MB total, ~7us @ 23.3 TB/s).
// 2-D grid: row = blockIdx.y, col = blockIdx.x*256+tid (no div/mod chains).
// ---------------------------------------------------------------------------
__global__ void k_find_col(const float* __restrict__ mat, int* __restrict__ out, int cols) {
    int col = blockIdx.x * blockDim.x + threadIdx.x;
    int row = blockIdx.y;
    if (col >= cols) return;
    if (mat[(size_t)row * cols + col] > 0.5f) out[row] = col;
}

__global__ void k_find_row(const float* __restrict__ mat, int* __restrict__ out, int cols) {
    int col = blockIdx.x * blockDim.x + threadIdx.x;
    int row = blockIdx.y;
    if (col >= cols) return;
    if (mat[(size_t)row * cols + col] > 0.5f) out[col] = row;
}

__global__ void k_zero_i(int* __restrict__ p, int n) {
    int i = blockIdx.x * blockDim.x + threadIdx.x;
    if (i < n) p[i] = 0;
}

// Pack (head, tail, rel) into one word: [10:0]=head, [21:11]=tail, [31:22]=rel
__global__ void k_pack(const int* __restrict__ heads, const int* __restrict__ tails,
                       const int* __restrict__ rels, unsigned* __restrict__ pk) {
    int n = blockIdx.x * blockDim.x + threadIdx.x;
    if (n >= N_) return;
    pk[n] = (unsigned)heads[n] | ((unsigned)tails[n] << 11) | ((unsigned)rels[n] << 22);
}

// cnt_er[e,r] = #{n : tail=e, rel=r}; cnt_bx[b,r] = #{n : head=ent[b], rel=r}
__global__ void k_count(const int* __restrict__ heads, const int* __restrict__ tails,
                        const int* __restrict__ rels, const int* __restrict__ ent,
                        int* __restrict__ cnt_er, int* __restrict__ cnt_bx) {
    int n = blockIdx.x * blockDim.x + threadIdx.x;
    if (n >= N_) return;
    int hh = heads[n], tl = tails[n], r = rels[n];
    atomicAdd(&cnt_er[tl * NREL_ + r], 1);
#pragma unroll
    for (int b = 0; b < B_; ++b)
        if (ent[b] == hh) atomicAdd(&cnt_bx[b * NREL_ + r], 1);
}

// hidden_base = clip01(e2r)[:, :24]
__global__ void k_hb(const int* __restrict__ cnt_er, float* __restrict__ hb) {
    int i = blockIdx.x * blockDim.x + threadIdx.x;
    if (i >= E_ * 24) return;
    int e = i / 24, j = i % 24;
    hb[i] = fminf((float)cnt_er[e * NREL_ + j], 1.0f);
}

// gate_x, hx_lin[b,l], hidden_type_x[b,l]  (single 256-thread block)
__global__ void k_prepx(const int* __restrict__ cnt_bx, const int* __restrict__ ent,
                        const int* __restrict__ tidx,
                        const float* __restrict__ h_x, const float* __restrict__ h_x_type,
                        const float* __restrict__ alpha_x, const float* __restrict__ beta_x,
                        float* __restrict__ hx_lin, float* __restrict__ htx,
                        float* __restrict__ gate_x) {
    __shared__ float hxr[B_][NREL_];
    __shared__ float hxb[B_][24];
    int tid = threadIdx.x;
    if (tid < B_ * NREL_) hxr[tid / NREL_][tid % NREL_] = fminf((float)cnt_bx[tid], 1.0f);
    __syncthreads();
    if (tid < B_ * 24) {
        int b = tid / 24, j = tid % 24;
        // concat([hx_raw[:, 12:-1], hx_raw[:, :12]])  (half = 12)
        hxb[b][j] = (j < 12) ? hxr[b][12 + j] : hxr[b][j - 12];
    }
    __syncthreads();
    int b = tid >> 5, l = tid & 31;
    float acc = 0.0f;
#pragma unroll
    for (int j = 0; j < 24; ++j) acc += hxb[b][j] * clip01(h_x[l * 24 + j] * 0.1f);
    hx_lin[tid] = acc;
    htx[tid]    = clip01(h_x_type[l * K_ + tidx[ent[b]]] * 0.1f);
    if (tid < L_)
        gate_x[tid] = 1.0f - clip01(clip01(alpha_x[tid] * 0.1f) + clip01(beta_x[tid] * 0.1f));
}

// hidden_x[b,l] = clip01(a0*htx + b0*hx_lin) + gate_x   (uses alpha[0], beta[0])
__global__ void k_hiddenx(const float* __restrict__ alpha0, const float* __restrict__ beta0,
                          const float* __restrict__ hx_lin, const float* __restrict__ htx,
                          const float* __restrict__ gate_x, float* __restrict__ hidden_x) {
    int tid = threadIdx.x;
    int l = tid & 31;
    float a = clip01(alpha0[l] * 0.1f), b = clip01(beta0[l] * 0.1f);
    hidden_x[tid] = clip01(a * htx[tid] + b * hx_lin[tid]) + gate_x[l];
}

// softmax(w[t], axis=-1), one thread per l
__global__ void k_wprobs(const float* __restrict__ wt, float* __restrict__ wp) {
    int l = threadIdx.x;
    if (l >= L_) return;
    float v[NREL_];
    float m = -1e30f;
    for (int r = 0; r < NREL_; ++r) { v[r] = wt[l * NREL_ + r]; m = fmaxf(m, v[r]); }
    float s = 0.0f;
    for (int r = 0; r < NREL_; ++r) { v[r] = __expf(v[r] - m); s += v[r]; }
    float inv = 1.0f / s;
    for (int r = 0; r < NREL_; ++r) wp[l * NREL_ + r] = v[r] * inv;
}

// ---------------------------------------------------------------------------
// hidden[l,e] = clip01( A[e,0:40] @ Bm[0:40,l] ) + (1 - clip01(a_l + b_l))
//   A[e,j]  = j<24 ? hidden_base[e,j]            : (tidx[e]==j-24)
//   Bm[j,l] = j<24 ? b_l*clip01(h[t][l,j]/10)    : a_l*clip01(h_type[t][l,j-24]/10)
// V_WMMA_F32_16X16X4_F32: 125 M-tiles x 2 N-tiles, 10 k-steps (K=40).
// ---------------------------------------------------------------------------
__global__ void k_hidden_wmma(const float* __restrict__ hb, const int* __restrict__ tidx,
                              const float* __restrict__ h_t, const float* __restrict__ htype_t,
                              const float* __restrict__ alpha_t, const float* __restrict__ beta_t,
                              float* __restrict__ hidden) {
    const int tile = blockIdx.x;     // 0..249
    const int tM = tile >> 1;        // e-block 0..124
    const int tN = tile & 1;         // l-block 0..1
    const int lane = threadIdx.x & 31;
    const int half = lane >> 4;
    const int lm = lane & 15;
    const int e = tM * 16 + lm;      // A row for this lane (always < 2000)
    const int l = tN * 16 + lm;      // B col for this lane

    const float a_l = clip01(alpha_t[l] * 0.1f);
    const float b_l = clip01(beta_t[l] * 0.1f);
    const int myT = tidx[e];

    v8f c;
#pragma unroll
    for (int i = 0; i < 8; ++i) c[i] = 0.0f;

#pragma unroll
    for (int ks = 0; ks < 10; ++ks) {
        const int k0 = ks * 4 + half * 2;
        v2f A, Bv;
#pragma unroll
        for (int q = 0; q < 2; ++q) {
            const int j = k0 + q;
            const float av = (j < 24) ? hb[e * 24 + j] : ((myT == (j - 24)) ? 1.0f : 0.0f);
            const float bv = (j < 24) ? (b_l * clip01(h_t[l * 24 + j] * 0.1f))
                                      : (a_l * clip01(htype_t[l * K_ + (j - 24)] * 0.1f));
            A[q]  = av;
            Bv[q] = bv;
        }
        c = __builtin_amdgcn_wmma_f32_16x16x4_f32(false, A, false, Bv, (short)0, c,
                                                  false, false);
    }

    const float gate = 1.0f - clip01(a_l + b_l);
#pragma unroll
    for (int v = 0; v < 8; ++v) {
        const int row = tM * 16 + half * 8 + v;   // e index of c[v]
        hidden[(size_t)l * E_ + row] = clip01(c[v]) + gate;   // column == this lane's l
    }
}

// s_prev[b,l,e] = (e == ent[b])
__global__ void k_init_s(const int* __restrict__ ent, float* __restrict__ s) {
    int i = blockIdx.x * blockDim.x + threadIdx.x;
    if (i >= B_ * L_ * E_) return;
    int e = i % E_;
    int b = i / (L_ * E_);
    s[i] = (e == ent[b]) ? 1.0f : 0.0f;
}

// ---------------------------------------------------------------------------
// One (b,l) channel per block. s_prev row staged into LDS with
// GLOBAL_LOAD_ASYNC_TO_LDS_B128 (ASYNCcnt path), scatter via ds_add_f32,
// result scaled in LDS and written back with GLOBAL_STORE_ASYNC_FROM_LDS_B128.
// s_next[b,l,e] = (sum_{n:tail=e} s_prev[b,l,head]*wp[l,rel]) * hidden[l,e] * (t==0 ? hidden_x[b,l] : 1)
// ---------------------------------------------------------------------------
#define ROW_XFERS ((E_ * 4) / 16)   // 500 x B128 transfers per 8000-byte row

__global__ void k_propagate(const float* __restrict__ s_prev, float* __restrict__ s_next,
                            const unsigned* __restrict__ pk,
                            const float* __restrict__ wp,
                            const float* __restrict__ hidden,
                            const float* __restrict__ hidden_x, int t) {
    __shared__ __align__(16) float sp[E_];
    __shared__ __align__(16) float sa[E_];
    __shared__ float wpl[NREL_];
    const int bl = blockIdx.x;        // 0..255 == b*32 + l
    const int l = bl & 31;
    const int tid = threadIdx.x;      // 0..255

    // Async-stage s_prev row (global -> LDS, no VGPR round-trip)
    const float* __restrict__ srow = s_prev + (size_t)bl * E_;
    {
        const unsigned spBase = (unsigned)(uintptr_t)&sp[0];
        const unsigned long long gbase = (unsigned long long)(uintptr_t)srow;
        for (int i = tid; i < ROW_XFERS; i += 256) {
            const unsigned off = (unsigned)i * 16u;
            asm volatile("global_load_async_to_lds_b128 %0, %1, %2"
                         :
                         : "v"(spBase + off), "v"(off), "s"(gbase)
                         : "memory");
        }
    }
    for (int e = tid; e < E_; e += 256) sa[e] = 0.0f;
    if (tid < NREL_) wpl[tid] = wp[l * NREL_ + tid];
    asm volatile("s_wait_asynccnt 0x0" ::: "memory");
    __syncthreads();

    for (int n = tid; n < N_; n += 256) {
        const unsigned wrd = pk[n];
        const float v = sp[wrd & 2047u];
        if (v != 0.0f) atomicAdd(&sa[(wrd >> 11) & 2047u], v * wpl[wrd >> 22]);
    }
    __syncthreads();

    const float hxm = (t == 0) ? hidden_x[bl] : 1.0f;
    const float* __restrict__ hrow = hidden + (size_t)l * E_;
    for (int e = tid; e < E_; e += 256) sa[e] *= hrow[e] * hxm;
    __syncthreads();

    // Async write-back (LDS -> global)
    {
        float* __restrict__ drow = s_next + (size_t)bl * E_;
        const unsigned saBase = (unsigned)(uintptr_t)&sa[0];
        const unsigned long long gbase = (unsigned long long)(uintptr_t)drow;
        for (int i = tid; i < ROW_XFERS; i += 256) {
            const unsigned off = (unsigned)i * 16u;
            asm volatile("global_store_async_from_lds_b128 %0, %1, %2"
                         :
                         : "v"(off), "v"(saBase + off), "s"(gbase)
                         : "memory");
        }
    }
    asm volatile("s_wait_asynccnt 0x0" ::: "memory");
}

// out[b,e] = sum_l s[b,l,e] * tanh(weight[l])
__global__ void k_out(const float* __restrict__ s, const float* __restrict__ weight,
                      float* __restrict__ out) {
    int i = blockIdx.x * blockDim.x + threadIdx.x;
    if (i >= B_ * E_) return;
    int b = i / E_, e = i % E_;
    float acc = 0.0f;
#pragma unroll
    for (int l = 0; l < L_; ++l) acc += s[((size_t)(b * L_ + l)) * E_ + e] * tanhf(weight[l]);
    out[i] = acc;
}

// ---------------------------------------------------------------------------
extern "C" void kernel_launch(void* const* d_in, const int* in_sizes, int n_in,
                              void* d_out, int out_size, void* d_ws, size_t ws_size,
                              hipStream_t stream) {
    const float* input_x  = (const float*)d_in[0];
    const float* type_mat = (const float*)d_in[1];
    const float* e2triple = (const float*)d_in[2];
    const float* triple2e = (const float*)d_in[3];
    const float* triple2r = (const float*)d_in[4];
    const float* w        = (const float*)d_in[5];
    const float* weight   = (const float*)d_in[6];
    const float* h        = (const float*)d_in[7];
    const float* h_x      = (const float*)d_in[8];
    const float* h_type   = (const float*)d_in[9];
    const float* h_x_type = (const float*)d_in[10];
    const float* alpha    = (const float*)d_in[11];
    const float* beta     = (const float*)d_in[12];
    const float* alpha_x  = (const float*)d_in[13];
    const float* beta_x   = (const float*)d_in[14];
    (void)in_sizes; (void)n_in; (void)out_size; (void)ws_size;

    // Workspace carve (all buffers fully (re)written every call)
    char* p = (char*)d_ws;
    auto carve = [&](size_t bytes) -> char* {
        char* r = p;
        p += (bytes + 255) & ~(size_t)255;
        return r;
    };
    int*      ent      = (int*)carve(B_ * sizeof(int));
    int*      tidx     = (int*)carve(E_ * sizeof(int));
    int*      heads    = (int*)carve(N_ * sizeof(int));
    int*      tails    = (int*)carve(N_ * sizeof(int));
    int*      rels     = (int*)carve(N_ * sizeof(int));
    unsigned* pk       = (unsigned*)carve(N_ * sizeof(unsigned));
    int*      cnt_er   = (int*)carve(E_ * NREL_ * sizeof(int));
    int*      cnt_bx   = (int*)carve(B_ * NREL_ * sizeof(int));
    float*    hb       = (float*)carve(E_ * 24 * sizeof(float));
    float*    hx_lin   = (float*)carve(B_ * L_ * sizeof(float));
    float*    htx      = (float*)carve(B_ * L_ * sizeof(float));
    float*    gate_x   = (float*)carve(L_ * sizeof(float));
    float*    hidden_x = (float*)carve(B_ * L_ * sizeof(float));
    float*    wp       = (float*)carve(L_ * NREL_ * sizeof(float));
    float*    hidden   = (float*)carve((size_t)L_ * E_ * sizeof(float));
    float*    s0       = (float*)carve((size_t)B_ * L_ * E_ * sizeof(float));
    float*    s1       = (float*)carve((size_t)B_ * L_ * E_ * sizeof(float));

    const int TPB = 256;
    auto grid1 = [](long long n, int tpb) { return (unsigned)((n + tpb - 1) / tpb); };

    // Zero atomic count buffers
    k_zero_i<<<grid1(E_ * NREL_, TPB), TPB, 0, stream>>>(cnt_er, E_ * NREL_);
    k_zero_i<<<grid1(B_ * NREL_, TPB), TPB, 0, stream>>>(cnt_bx, B_ * NREL_);

    // Index extraction from one-hot matrices (2-D grids: y = row)
    k_find_col<<<dim3(grid1(E_, TPB), B_), TPB, 0, stream>>>(input_x, ent, E_);
    k_find_col<<<dim3(grid1(K_, TPB), E_), TPB, 0, stream>>>(type_mat, tidx, K_);
    k_find_col<<<dim3(grid1(NREL_, TPB), N_), TPB, 0, stream>>>(triple2r, rels, NREL_);
    k_find_col<<<dim3(grid1(E_, TPB), N_), TPB, 0, stream>>>(triple2e, tails, E_);
    k_find_row<<<dim3(grid1(N_, TPB), E_), TPB, 0, stream>>>(e2triple, heads, N_);

    // Pack triples; degree counts; small precomputations
    k_pack<<<grid1(N_, TPB), TPB, 0, stream>>>(heads, tails, rels, pk);
    k_count<<<grid1(N_, TPB), TPB, 0, stream>>>(heads, tails, rels, ent, cnt_er, cnt_bx);
    k_hb<<<grid1(E_ * 24, TPB), TPB, 0, stream>>>(cnt_er, hb);
    k_prepx<<<1, 256, 0, stream>>>(cnt_bx, ent, tidx, h_x, h_x_type, alpha_x, beta_x,
                                   hx_lin, htx, gate_x);
    k_hiddenx<<<1, 256, 0, stream>>>(alpha, beta, hx_lin, htx, gate_x, hidden_x);

    // s_prev init (t=0 effective input is onehot(ent[b]))
    k_init_s<<<grid1((long long)B_ * L_ * E_, TPB), TPB, 0, stream>>>(ent, s0);

    float* s_cur = s0;
    float* s_nxt = s1;
    for (int t = 0; t < T_; ++t) {
        k_wprobs<<<1, 32, 0, stream>>>(w + (size_t)t * L_ * NREL_, wp);
        k_hidden_wmma<<<250, 32, 0, stream>>>(hb, tidx,
                                              h + (size_t)t * L_ * 24,
                                              h_type + (size_t)t * L_ * K_,
                                              alpha + (size_t)t * L_,
                                              beta + (size_t)t * L_, hidden);
        k_propagate<<<B_ * L_, 256, 0, stream>>>(s_cur, s_nxt, pk, wp, hidden, hidden_x, t);
        float* tmp = s_cur; s_cur = s_nxt; s_nxt = tmp;
    }

    k_out<<<grid1((long long)B_ * E_, TPB), TPB, 0, stream>>>(s_cur, weight, (float*)d_out);
}